// MaskedDiffusionModule_68942815035566
// MI455X (gfx1250) — compile-verified
//
#include <hip/hip_runtime.h>
#include <hip/hip_bf16.h>

// ---------------------------------------------------------------------------
// MaskedDiffusionLoss for MI455X (gfx1250, wave32).
// Memory-bound: ~0.53 GB traffic -> ~23 us floor at 23.3 TB/s.
// Kernel 1: one block per (b,n) row; LDS-resident bucket select for the exact
//           1200th-smallest rand value, fused with mask/token substitution and
//           masked-MSE partial reduction.
// Kernel 2: tiny position masking + partials.
// Kernel 3: single-wave WMMA (v_wmma_f32_16x16x4_f32, B = ones) reduction of
//           all partials -> scalar loss.
// ---------------------------------------------------------------------------

#define BB   16
#define NN   1024
#define GG   2000
#define ROWS (BB * NN)          // 16384
#define BLK  256
#define BINS 2048
#define RANK 1199               // 0-based index of kth smallest (k = G - 800 = 1200)
#define CAP  256

typedef __attribute__((ext_vector_type(2))) float v2f;
typedef __attribute__((ext_vector_type(8))) float v8f;

// ------------------------- expression kernel -------------------------------
__global__ __launch_bounds__(BLK) void expr_kernel(
    const float* __restrict__ expr,
    const float* __restrict__ pred,
    const float* __restrict__ rnd,
    const float* __restrict__ token,     // (G,)
    float* __restrict__ out_masked,
    float* __restrict__ rsum,            // (ROWS,)
    float* __restrict__ rcnt)            // (ROWS,)
{
    __shared__ float    vals[GG];
    __shared__ unsigned hist[BINS];
    __shared__ unsigned scanbuf[BLK];
    __shared__ float    cand[CAP];
    __shared__ unsigned candCount;
    __shared__ unsigned s_bucket;
    __shared__ unsigned s_rlocal;
    __shared__ float    s_thr;
    __shared__ float    redS[BLK];
    __shared__ unsigned redC[BLK];

    const int tid = threadIdx.x;
    const long long base = (long long)blockIdx.x * GG;

    for (int i = tid; i < BINS; i += BLK) hist[i] = 0u;
    if (tid == 0) { candCount = 0u; s_thr = 0.0f; }
    __syncthreads();

    // Load rand row to LDS + build 2048-bin histogram.
    // bucket = floor(v * 2048): exact (power-of-two scale), monotone in v.
    for (int i = tid; i < GG; i += BLK) {
        float v = rnd[base + i];
        vals[i] = v;
        int bkt = (int)(v * 2048.0f);
        bkt = bkt < 0 ? 0 : (bkt > BINS - 1 ? BINS - 1 : bkt);
        atomicAdd(&hist[bkt], 1u);
    }
    __syncthreads();

    // Chunk sums: 8 bins per thread, then Hillis-Steele inclusive scan.
    unsigned cs = 0u;
#pragma unroll
    for (int j = 0; j < 8; ++j) cs += hist[tid * 8 + j];
    unsigned acc = cs;
    scanbuf[tid] = acc;
    __syncthreads();
    for (int off = 1; off < BLK; off <<= 1) {
        unsigned t = (tid >= off) ? scanbuf[tid - off] : 0u;
        __syncthreads();
        acc += t;
        scanbuf[tid] = acc;
        __syncthreads();
    }
    const unsigned S = acc;          // inclusive
    const unsigned E = S - cs;       // exclusive

    // Exactly one thread owns the chunk containing global rank RANK.
    if (E <= (unsigned)RANK && (unsigned)RANK < S) {
        unsigned running = E;
#pragma unroll
        for (int j = 0; j < 8; ++j) {
            unsigned c = hist[tid * 8 + j];
            if (running + c > (unsigned)RANK) {
                s_bucket = (unsigned)(tid * 8 + j);
                s_rlocal = (unsigned)RANK - running;
                break;
            }
            running += c;
        }
    }
    __syncthreads();
    const unsigned tb = s_bucket;
    const unsigned rl = s_rlocal;

    // Gather candidates in the target bucket (expected ~1-3 elements).
    for (int i = tid; i < GG; i += BLK) {
        float vv = vals[i];
        int bkt = (int)(vv * 2048.0f);
        bkt = bkt < 0 ? 0 : (bkt > BINS - 1 ? BINS - 1 : bkt);
        if ((unsigned)bkt == tb) {
            unsigned p = atomicAdd(&candCount, 1u);
            if (p < CAP) cand[p] = vv;
        }
    }
    __syncthreads();

    // Exact counting select within the bucket (tie-safe: duplicates handled
    // via equality count; the selected value is unique & order-independent).
    unsigned c = candCount; if (c > CAP) c = CAP;
    if ((unsigned)tid < c) {
        float x = cand[tid];
        unsigned less = 0u, eq = 0u;
        for (unsigned j = 0; j < c; ++j) {
            float y = cand[j];
            less += (y < x) ? 1u : 0u;
            eq   += (y == x) ? 1u : 0u;
        }
        if (less <= rl && rl < less + eq) s_thr = x;
    }
    __syncthreads();
    const float thr = s_thr;

    // Fused elementwise pass: token substitution + masked squared-error.
    float    fs = 0.0f;
    unsigned fc = 0u;
    for (int i = tid; i < GG; i += BLK) {
        __builtin_prefetch(expr + base + i + 2 * BLK, 0, 0);   // global_prefetch_b8
        __builtin_prefetch(pred + base + i + 2 * BLK, 0, 0);
        float r = vals[i];
        bool  m = (r >= thr);
        float e = expr[base + i];
        float p = pred[base + i];
        out_masked[base + i] = m ? token[i] : e;
        if (m) { float d = p - e; fs += d * d; fc++; }
    }

    // Deterministic block tree reduction.
    redS[tid] = fs;
    redC[tid] = fc;
    __syncthreads();
    for (int off = BLK / 2; off > 0; off >>= 1) {
        if (tid < off) { redS[tid] += redS[tid + off]; redC[tid] += redC[tid + off]; }
        __syncthreads();
    }
    if (tid == 0) {
        rsum[blockIdx.x] = redS[0];
        rcnt[blockIdx.x] = (float)redC[0];
    }
}

// --------------------------- position kernel -------------------------------
__global__ __launch_bounds__(BLK) void pos_kernel(
    const float* __restrict__ pos,
    const float* __restrict__ predp,
    const float* __restrict__ rp,
    const float* __restrict__ ptoken,    // (3,)
    const int*   __restrict__ dimi,
    const int*   __restrict__ exci,
    float* __restrict__ out_pos,
    float* __restrict__ psum,            // (64,)
    float* __restrict__ pcnt)            // (64,)
{
    const int tid = threadIdx.x;
    const int row = blockIdx.x * BLK + tid;      // 64 * 256 == ROWS exactly

    float    fs = 0.0f;
    unsigned fc = 0u;
    {
        const float M2 = (float)(0.33 * 0.3);    // matches POS_RATIO*0.3 in f32
        const float M1 = 0.33f;
        float r  = rp[row];
        bool  m2 = r < M2;
        bool  m1 = (r < M1) && !m2;
        int   di = dimi[row];
        int   ei = exci[row];
#pragma unroll
        for (int d = 0; d < 3; ++d) {
            bool  m = (m1 && d == di) || (m2 && d != ei);
            float p = pos[row * 3 + d];
            out_pos[row * 3 + d] = m ? ptoken[d] : p;
            if (m) { float dd = predp[row * 3 + d] - p; fs += dd * dd; fc++; }
        }
    }

    __shared__ float    rs[BLK];
    __shared__ unsigned rc[BLK];
    rs[tid] = fs; rc[tid] = fc;
    __syncthreads();
    for (int off = BLK / 2; off > 0; off >>= 1) {
        if (tid < off) { rs[tid] += rs[tid + off]; rc[tid] += rc[tid + off]; }
        __syncthreads();
    }
    if (tid == 0) {
        psum[blockIdx.x] = rs[0];
        pcnt[blockIdx.x] = (float)rc[0];
    }
}

// ------------------------ final WMMA reduction -----------------------------
// Sum n (multiple of 64) floats with V_WMMA_F32_16X16X4_F32, B = ones:
// D[m][n] = sum_k A[m][k] + C[m][n]; any bijection of 64 values onto (m,k)
// works for a total sum. Whole wave (32 lanes) participates, EXEC all-ones.
__device__ __forceinline__ float wmma_sum64(const float* __restrict__ buf,
                                            int n, int lane)
{
    v8f c = {};
    v2f b; b.x = 1.0f; b.y = 1.0f;
    for (int i = 0; i < n; i += 64) {
        v2f a;
        a.x = buf[i + lane];
        a.y = buf[i + 32 + lane];
        c = __builtin_amdgcn_wmma_f32_16x16x4_f32(
                /*neg_a=*/false, a, /*neg_b=*/false, b,
                /*c_mod=*/(short)0, c, /*reuse_a=*/false, /*reuse_b=*/false);
    }
    // D columns are identical; lane0 component-sum covers M=0..7,
    // lane16 covers M=8..15.
    float s = c[0] + c[1] + c[2] + c[3] + c[4] + c[5] + c[6] + c[7];
    return __shfl(s, 0, 32) + __shfl(s, 16, 32);
}

__global__ __launch_bounds__(32) void final_kernel(
    const float* __restrict__ rsum, const float* __restrict__ rcnt,
    const float* __restrict__ psum, const float* __restrict__ pcnt,
    float* __restrict__ out_loss)
{
    const int lane = threadIdx.x;
    float se = wmma_sum64(rsum, ROWS, lane);
    float ce = wmma_sum64(rcnt, ROWS, lane);
    float sp = wmma_sum64(psum, 64, lane);
    float cp = wmma_sum64(pcnt, 64, lane);
    if (lane == 0) {
        *out_loss = se / fmaxf(ce, 1.0f) + sp / fmaxf(cp, 1.0f);
    }
}

// ------------------------------ launcher -----------------------------------
extern "C" void kernel_launch(void* const* d_in, const int* in_sizes, int n_in,
                              void* d_out, int out_size, void* d_ws, size_t ws_size,
                              hipStream_t stream)
{
    const float* expr  = (const float*)d_in[0];   // (B,N,G)
    const float* pos   = (const float*)d_in[1];   // (B,N,3)
    const float* pexpr = (const float*)d_in[2];   // (B,N,G)
    const float* ppos  = (const float*)d_in[3];   // (B,N,3)
    const float* rexpr = (const float*)d_in[4];   // (B,N,G)
    const float* rpos  = (const float*)d_in[5];   // (B,N)
    const float* etok  = (const float*)d_in[6];   // (1,1,G)
    const float* ptok  = (const float*)d_in[7];   // (1,1,3)
    const int*   dimi  = (const int*)d_in[8];     // (B,N)
    const int*   exci  = (const int*)d_in[9];     // (B,N)

    float* out_mexpr = (float*)d_out;
    float* out_mpos  = out_mexpr + (size_t)ROWS * GG;
    float* out_loss  = out_mpos  + (size_t)ROWS * 3;

    float* ws   = (float*)d_ws;
    float* rsum = ws;                 // ROWS
    float* rcnt = ws + ROWS;          // ROWS
    float* psum = ws + 2 * ROWS;      // 64
    float* pcnt = ws + 2 * ROWS + 64; // 64

    expr_kernel<<<ROWS, BLK, 0, stream>>>(expr, pexpr, rexpr, etok,
                                          out_mexpr, rsum, rcnt);
    pos_kernel<<<ROWS / BLK, BLK, 0, stream>>>(pos, ppos, rpos, ptok, dimi, exci,
                                               out_mpos, psum, pcnt);
    final_kernel<<<1, 32, 0, stream>>>(rsum, rcnt, psum, pcnt, out_loss);
}